// KNNLearner_62620623175912
// MI455X (gfx1250) — compile-verified
//
#include <hip/hip_runtime.h>
#include <hip/hip_bf16.h>

#define NS 16384
#define NQ 16384
#define DIM 64
#define NCLS 64
#define SPLIT 4
#define TILES_PER_SPLIT (NS / 16 / SPLIT)   // 256 tiles of 16 supports per split
#define LDS_ROW 68                          // 64 + 4 pad floats: conflict-free b64 reads

typedef float v2f __attribute__((ext_vector_type(2)));
typedef float v8f __attribute__((ext_vector_type(8)));

// ---------------------------------------------------------------------------
// Kernel 1: per-support ||s||^2 and integer label (first argmax of one-hot)
// ---------------------------------------------------------------------------
__global__ __launch_bounds__(256) void knn_prep(const float* __restrict__ S,
                                                const float* __restrict__ onehot,
                                                float* __restrict__ sqs,
                                                int* __restrict__ labels) {
    int s = blockIdx.x * blockDim.x + threadIdx.x;
    if (s >= NS) return;
    const float* r = S + (size_t)s * DIM;
    float acc = 0.0f;
#pragma unroll
    for (int d = 0; d < DIM; ++d) acc = fmaf(r[d], r[d], acc);
    sqs[s] = acc;

    const float* lr = onehot + (size_t)s * NCLS;
    float best = lr[0];
    int bi = 0;
#pragma unroll 4
    for (int j = 1; j < NCLS; ++j) {
        float v = lr[j];
        if (v > best) { best = v; bi = j; }   // strict > keeps first argmax
    }
    labels[s] = bi;
}

// ---------------------------------------------------------------------------
// Kernel 2: fused GEMM (f32 WMMA) + running argmin over a support split.
// Block = 256 threads = 8 waves; each wave owns a 16-query tile.
// Support tiles (16 x 64 f32 = 4KB) staged in LDS, double buffered.
// ---------------------------------------------------------------------------
__global__ __launch_bounds__(256) void knn_main(const float* __restrict__ Q,
                                                const float* __restrict__ S,
                                                const float* __restrict__ sqs,
                                                float* __restrict__ pval,
                                                int* __restrict__ pidx) {
    __shared__ float lds[2][16 * LDS_ROW];

    const int tid   = threadIdx.x;
    const int wave  = tid >> 5;
    const int lane  = tid & 31;
    const int h     = lane >> 4;   // half-wave (selects K pair / M offset)
    const int g     = lane & 15;   // position within half (N column / M row)
    const int split = blockIdx.y;
    const int q0    = blockIdx.x * 128 + wave * 16;

    // --- Load A fragments once: a[k] holds Q[q0+g][4k + 2h + {0,1}] ---
    v2f a[16];
    {
        const float* qbase = Q + (size_t)(q0 + g) * DIM + 2 * h;
#pragma unroll
        for (int k = 0; k < 16; ++k)
            a[k] = *(const v2f*)(qbase + 4 * k);
    }

    float bestv[8];
    int   besti[8];
#pragma unroll
    for (int v = 0; v < 8; ++v) { bestv[v] = 3.4e38f; besti[v] = 0x7fffffff; }

    // cooperative staging map: 256 threads x float4 = one 16x64 tile
    const int srow = tid >> 4;          // 0..15
    const int scol = (tid & 15) * 4;    // 0..60

    const int t0 = split * TILES_PER_SPLIT;
    float4 pre = *(const float4*)(S + (size_t)(t0 * 16 + srow) * DIM + scol);

    int buf = 0;
    for (int tt = 0; tt < TILES_PER_SPLIT; ++tt) {
        const int t = t0 + tt;

        *(float4*)(&lds[buf][srow * LDS_ROW + scol]) = pre;
        __syncthreads();

        if (tt + 1 < TILES_PER_SPLIT) {
            pre = *(const float4*)(S + (size_t)((t + 1) * 16 + srow) * DIM + scol);
            if (tt + 2 < TILES_PER_SPLIT)
                __builtin_prefetch(S + (size_t)((t + 2) * 16 + srow) * DIM + scol, 0, 0);
        }

        const float sq   = sqs[t * 16 + g];
        const int   sidx = t * 16 + g;

        // --- 16x16 dot-product tile: 16 chained f32 WMMAs over K=64 ---
        v8f c = {0.f, 0.f, 0.f, 0.f, 0.f, 0.f, 0.f, 0.f};
        const float* bbase = &lds[buf][g * LDS_ROW + 2 * h];
#pragma unroll
        for (int k = 0; k < 16; ++k) {
            v2f b = *(const v2f*)(bbase + 4 * k);
            c = __builtin_amdgcn_wmma_f32_16x16x4_f32(
                    /*neg_a=*/false, a[k], /*neg_b=*/false, b,
                    /*c_mod=*/(short)0, c, /*reuse_a=*/false, /*reuse_b=*/false);
        }

        // lane holds dot(q = v + 8h, s = sidx) in c[v]
#pragma unroll
        for (int v = 0; v < 8; ++v) {
            float val = fmaf(-2.0f, c[v], sq);     // ||s||^2 - 2 q.s
            if (val < bestv[v]) { bestv[v] = val; besti[v] = sidx; }
        }
        __syncthreads();
        buf ^= 1;
    }

    // --- reduce min/argmin across the 16 lanes of each half-wave ---
#pragma unroll
    for (int v = 0; v < 8; ++v) {
        float bv = bestv[v];
        int   bi = besti[v];
#pragma unroll
        for (int off = 8; off >= 1; off >>= 1) {
            float ov = __shfl_xor(bv, off, 16);
            int   oi = __shfl_xor(bi, off, 16);
            if (ov < bv || (ov == bv && oi < bi)) { bv = ov; bi = oi; }
        }
        if (g == 0) {
            int qi = q0 + v + 8 * h;               // query row for C VGPR v, half h
            pval[(size_t)split * NQ + qi] = bv;
            pidx[(size_t)split * NQ + qi] = bi;
        }
    }
}

// ---------------------------------------------------------------------------
// Kernel 3: combine SPLIT partials (ascending split order => first-argmin
// tie-break preserved), gather labels, write one-hot rows coalesced.
// ---------------------------------------------------------------------------
__global__ __launch_bounds__(256) void knn_combine(const float* __restrict__ pval,
                                                   const int* __restrict__ pidx,
                                                   const int* __restrict__ labels,
                                                   float* __restrict__ out) {
    __shared__ int lab_sm[256];
    const int tid = threadIdx.x;
    const int q   = blockIdx.x * 256 + tid;

    float bv = 3.4e38f;
    int   bi = 0;
#pragma unroll
    for (int p = 0; p < SPLIT; ++p) {
        float v = pval[(size_t)p * NQ + q];
        if (v < bv) { bv = v; bi = pidx[(size_t)p * NQ + q]; }
    }
    lab_sm[tid] = labels[bi];
    __syncthreads();

    const size_t base = (size_t)blockIdx.x * 256 * NCLS;
#pragma unroll 4
    for (int i = tid; i < 256 * NCLS; i += 256) {
        int ql = i >> 6;          // local query
        int j  = i & 63;          // class
        out[base + i] = (j == lab_sm[ql]) ? 1.0f : 0.0f;
    }
}

// ---------------------------------------------------------------------------
extern "C" void kernel_launch(void* const* d_in, const int* in_sizes, int n_in,
                              void* d_out, int out_size, void* d_ws, size_t ws_size,
                              hipStream_t stream) {
    const float* S      = (const float*)d_in[0];   // support_embeddings [NS, D]
    const float* Q      = (const float*)d_in[1];   // query_embeddings   [NQ, D]
    const float* onehot = (const float*)d_in[2];   // support_labels_onehot [NS, 64]
    float* out = (float*)d_out;

    // workspace layout
    char* ws = (char*)d_ws;
    float* sqs    = (float*)ws;                                  // NS floats
    int*   labels = (int*)(ws + NS * sizeof(float));             // NS ints
    float* pval   = (float*)(ws + NS * 8);                       // SPLIT*NQ floats
    int*   pidx   = (int*)(ws + NS * 8 + (size_t)SPLIT * NQ * 4);// SPLIT*NQ ints

    knn_prep<<<(NS + 255) / 256, 256, 0, stream>>>(S, onehot, sqs, labels);

    dim3 grid(NQ / 128, SPLIT);
    knn_main<<<grid, 256, 0, stream>>>(Q, S, sqs, pval, pidx);

    knn_combine<<<NQ / 256, 256, 0, stream>>>(pval, pidx, labels, out);
}